// _YOLO_84361747628425
// MI455X (gfx1250) — compile-verified
//
#include <hip/hip_runtime.h>
#include <math.h>

#define NIMG 128
#define HH 26
#define WW 26
#define AA 5
#define CC 80
#define MM 16
#define HWC (HH * WW) /* 676 */

typedef __attribute__((ext_vector_type(2))) float v2f;
typedef __attribute__((ext_vector_type(8))) float v8f;

// ---------------------------------------------------------------------------
// Kernel 1: one block per image. 256 threads = 8 wave32s.
//   Phase A: all threads stream pred_o for the background objectness term.
//   Phase B: threads 0..15 each own one GT box (gather path: 5 anchor boxes,
//            one 80-float score row -> prefetched).
//   Deterministic in-block tree reduction -> partial[3][NIMG] in workspace.
// ---------------------------------------------------------------------------
__global__ __launch_bounds__(256) void yolo_img(
    const float* __restrict__ pb,   // (N,HW,A,4)
    const float* __restrict__ po,   // (N,HW,A)
    const float* __restrict__ ps,   // (N,HW,A,C)
    const float* __restrict__ gtb,  // (N,M,4)
    const int*   __restrict__ gtl,  // (N,M)
    float* __restrict__ partial)    // (3,NIMG)
{
  const int img = blockIdx.x;
  const int tid = threadIdx.x;

  const float* pbI = pb  + (size_t)img * HWC * AA * 4;
  const float* poI = po  + (size_t)img * HWC * AA;
  const float* psI = ps  + (size_t)img * HWC * AA * CC;
  const float* gI  = gtb + (size_t)img * MM * 4;
  const int*   lI  = gtl + (size_t)img * MM;

  float obj = 0.0f, bbox = 0.0f, clf = 0.0f;

  // -------- Phase A: background objectness over every cell ----------------
  for (int c = tid; c < HWC; c += 256) {
    const float* p = poI + c * AA;
    float mx = fmaxf(fmaxf(fmaxf(p[0], p[1]), fmaxf(p[2], p[3])), p[4]);
    obj += 0.5f * mx * mx;
  }

  // -------- Phase B: one GT box per thread (tid < 16) ---------------------
  if (tid < MM) {
    const float g0 = gI[tid * 4 + 0];
    const float g1 = gI[tid * 4 + 1];
    const float g2 = gI[tid * 4 + 2];
    const float g3 = gI[tid * 4 + 3];

    // cell index: int32 truncation of ((g0+g2)*W)/2 exactly as the reference
    const int cx = (int)(((g0 + g2) * (float)WW) * 0.5f);
    const int cy = (int)(((g1 + g3) * (float)HH) * 0.5f);
    const int cell = cy * WW + cx;

    // warm L2/WGP$ for the score rows we will gather below
    __builtin_prefetch(psI + (size_t)cell * AA * CC, 0, 1);

    const float area_b = (g2 - g0) * (g3 - g1);

    // IoU argmax over A=5 anchors (first-max tie policy via strict '>')
    int   besta = 0;
    float bestv = -1.0f;
    float s_lox = 0.f, s_loy = 0.f, s_hix = 0.f, s_hiy = 0.f;
    const float* cb = pbI + (size_t)cell * AA * 4;
#pragma unroll
    for (int a = 0; a < AA; ++a) {
      const float px = cb[a * 4 + 0], py = cb[a * 4 + 1];
      const float pw = cb[a * 4 + 2], ph = cb[a * 4 + 3];
      const float hix = px + pw, hiy = py + ph;          // hi = xy + wh
      const float lox = px - hix * 0.5f;                 // lo = xy - hi/2
      const float loy = py - hiy * 0.5f;
      const float area_a = (hix - lox) * (hiy - loy);
      const float ltx = fmaxf(lox, g0), lty = fmaxf(loy, g1);
      const float rbx = fminf(hix, g2), rby = fminf(hiy, g3);
      const float wx = fmaxf(rbx - ltx, 0.0f);
      const float wy = fmaxf(rby - lty, 0.0f);
      const float inter = wx * wy;
      const float iou = inter / (area_a + area_b - inter);
      if (iou > bestv) {
        bestv = iou; besta = a;
        s_lox = lox; s_loy = loy; s_hix = hix; s_hiy = hiy;
      }
    }

    // objectness at the selected cell: replace background term with residual
    const float* pc = poI + cell * AA;
    const float omax = fmaxf(fmaxf(fmaxf(pc[0], pc[1]), fmaxf(pc[2], pc[3])), pc[4]);
    const float sel_o = pc[besta];
    obj += (sel_o - bestv) * (sel_o - bestv) - 0.5f * omax * omax;

    // bbox term: cell-fraction offsets + sqrt-wh residuals
    const float gcx = (g0 + g2) * 0.5f * (float)WW;
    const float gcy = (g1 + g3) * 0.5f * (float)HH;
    const float gfx = gcx - floorf(gcx);
    const float gfy = gcy - floorf(gcy);
    const float sbx = s_lox * (float)WW;
    const float sby = s_loy * (float)HH;
    const float sfx = sbx - floorf(sbx);
    const float sfy = sby - floorf(sby);
    const float dw = sqrtf(s_hix) - sqrtf(g2 - g0);
    const float dh = sqrtf(s_hiy) - sqrtf(g3 - g1);
    bbox += (sfx - gfx) * (sfx - gfx) + (sfy - gfy) * (sfy - gfy) + dw * dw + dh * dh;

    // classification NLL via shifted log-softmax over C=80
    const float* sc = psI + ((size_t)cell * AA + besta) * CC;
    float mx = sc[0];
    for (int j = 1; j < CC; ++j) mx = fmaxf(mx, sc[j]);
    float se = 0.0f;
    for (int j = 0; j < CC; ++j) se += expf(sc[j] - mx);
    const int lab = lI[tid];
    clf += mx + logf(se) - sc[lab];
  }

  // -------- deterministic block reduction ---------------------------------
  __shared__ float s0[256], s1[256], s2[256];
  s0[tid] = obj; s1[tid] = bbox; s2[tid] = clf;
  __syncthreads();
#pragma unroll
  for (int s = 128; s > 0; s >>= 1) {
    if (tid < s) {
      s0[tid] += s0[tid + s];
      s1[tid] += s1[tid + s];
      s2[tid] += s2[tid + s];
    }
    __syncthreads();
  }
  if (tid == 0) {
    partial[0 * NIMG + img] = s0[0];
    partial[1 * NIMG + img] = s1[0];
    partial[2 * NIMG + img] = s2[0];
  }
}

// ---------------------------------------------------------------------------
// Kernel 2: single wave32. Reduce partial[3][128] -> out[3] with 32 chained
// V_WMMA_F32_16X16X4_F32 accumulations: A rows 0..2 carry the three loss
// components (4 images per step), B = ones, D accumulates D->C (the designed
// WMMA chaining path). Rows >= 3 are zero-masked by multiply so EXEC stays
// all-ones as WMMA requires. Fixed order => bit-deterministic across replays.
// ---------------------------------------------------------------------------
__global__ __launch_bounds__(32) void yolo_final(
    const float* __restrict__ partial, float* __restrict__ out)
{
#if __has_builtin(__builtin_amdgcn_wmma_f32_16x16x4_f32)
  const int lane = threadIdx.x;                  // 0..31
  const int m    = lane & 15;                    // A-matrix row
  const int kb   = (lane >> 4) << 1;             // lanes 0-15 -> K 0,1 ; lanes 16-31 -> K 2,3
  const float mask = (m < 3) ? 1.0f : 0.0f;
  const int   row  = (m < 3) ? m : 0;            // clamp: all lanes load valid addrs
  const float* src = partial + (size_t)row * NIMG;

  v8f acc = {0.f, 0.f, 0.f, 0.f, 0.f, 0.f, 0.f, 0.f};
  v2f bones; bones.x = 1.0f; bones.y = 1.0f;     // B = all-ones (layout-invariant)

  for (int t = 0; t < NIMG / 4; ++t) {
    v2f a;
    a.x = src[4 * t + kb + 0] * mask;            // A[m][kb]   (VGPR0 per 16x4 f32 layout)
    a.y = src[4 * t + kb + 1] * mask;            // A[m][kb+1] (VGPR1)
    acc = __builtin_amdgcn_wmma_f32_16x16x4_f32(
        /*neg_a=*/false, a, /*neg_b=*/false, bones,
        /*c_mod=*/(short)0, acc, /*reuse_a=*/false, /*reuse_b=*/false);
  }
  // C/D layout: VGPR j, lanes 0-15 => M=j, N=lane. Lane 0 holds D[0..2][0].
  if (lane == 0) {
    const float invn = 1.0f / (float)NIMG;
    out[0] = acc[0] * invn;   // obj
    out[1] = acc[1] * invn;   // bbox
    out[2] = acc[2] * invn;   // clf
  }
#else
  if (threadIdx.x == 0) {
    const float invn = 1.0f / (float)NIMG;
    for (int k = 0; k < 3; ++k) {
      float s = 0.0f;
      for (int i = 0; i < NIMG; ++i) s += partial[k * NIMG + i];
      out[k] = s * invn;
    }
  }
#endif
}

extern "C" void kernel_launch(void* const* d_in, const int* in_sizes, int n_in,
                              void* d_out, int out_size, void* d_ws, size_t ws_size,
                              hipStream_t stream) {
  const float* pred_boxes  = (const float*)d_in[0];  // (128,26,26,5,4)
  const float* pred_o      = (const float*)d_in[1];  // (128,26,26,5)
  const float* pred_scores = (const float*)d_in[2];  // (128,26,26,5,80)
  const float* gt_boxes    = (const float*)d_in[3];  // (128,16,4)
  const int*   gt_labels   = (const int*)d_in[4];    // (128,16)
  float* out     = (float*)d_out;                    // 3 floats
  float* partial = (float*)d_ws;                     // 3*128 floats of scratch

  yolo_img<<<NIMG, 256, 0, stream>>>(pred_boxes, pred_o, pred_scores,
                                     gt_boxes, gt_labels, partial);
  yolo_final<<<1, 32, 0, stream>>>(partial, out);
}